// LSSBlockTS_88519275970823
// MI455X (gfx1250) — compile-verified
//
#include <hip/hip_runtime.h>
#include <math.h>

// ---------------- model constants ----------------
#define BB    2
#define LL    2048
#define DM    256
#define DI    512
#define DS    16
#define DTRK  16
#define MROWS (BB * LL)   // 4096
#define EPSV  1e-5f

typedef float v2f __attribute__((ext_vector_type(2)));
typedef float v8f __attribute__((ext_vector_type(8)));

__device__ __forceinline__ float sigf(float x) { return 1.f / (1.f + expf(-x)); }

// =====================================================================
// fp32 WMMA GEMM with register blocking:
//   Y[M,N] = act( X[M,K] @ W[N,K]^T + bias ) + resid
// Each wave computes an (MT*16) x (NT*16) output block via
// V_WMMA_F32_16X16X4_F32, reusing A fragments NT times and B fragments
// MT times. act: 0 = none, 1 = gelu(exact), 2 = softplus
// =====================================================================
template <int MT, int NT>
__global__ void gemm_wmma_blk(const float* __restrict__ X, int lda,
                              const float* __restrict__ W, int ldw,
                              const float* __restrict__ bias,
                              const float* __restrict__ resid, int ldr,
                              float* __restrict__ Y, int ldc, int ycol0,
                              int K, int act)
{
    const int lane = threadIdx.x;      // blockDim.x == 32
    const int half = lane >> 4;        // 0 | 1
    const int lid  = lane & 15;
    const int m0 = blockIdx.y * (16 * MT);
    const int n0 = blockIdx.x * (16 * NT);

    // A: 16x4 f32; VGPR0 holds K=0 (lanes 0-15) / K=2 (lanes 16-31), VGPR1: K=1/K=3
    const float* Xr[MT];
#pragma unroll
    for (int mi = 0; mi < MT; ++mi)
        Xr[mi] = X + (size_t)(m0 + mi * 16 + lid) * lda + 2 * half;
    // B[k][n] = W[n][k]; same K split across lane halves
    const float* Wr[NT];
#pragma unroll
    for (int ni = 0; ni < NT; ++ni)
        Wr[ni] = W + (size_t)(n0 + ni * 16 + lid) * ldw + 2 * half;

    v8f c[MT][NT];
#pragma unroll
    for (int mi = 0; mi < MT; ++mi)
#pragma unroll
        for (int ni = 0; ni < NT; ++ni)
            c[mi][ni] = (v8f){0.f, 0.f, 0.f, 0.f, 0.f, 0.f, 0.f, 0.f};

    for (int k = 0; k < K; k += 4) {
        if ((k & 31) == 0 && k + 32 < K) {   // uniform branch: prefetch next line
#pragma unroll
            for (int mi = 0; mi < MT; ++mi)
                __builtin_prefetch((const void*)(Xr[mi] + k + 32), 0, 1);
#pragma unroll
            for (int ni = 0; ni < NT; ++ni)
                __builtin_prefetch((const void*)(Wr[ni] + k + 32), 0, 1);
        }
        v2f a[MT], b[NT];
#pragma unroll
        for (int mi = 0; mi < MT; ++mi) { a[mi][0] = Xr[mi][k]; a[mi][1] = Xr[mi][k + 1]; }
#pragma unroll
        for (int ni = 0; ni < NT; ++ni) { b[ni][0] = Wr[ni][k]; b[ni][1] = Wr[ni][k + 1]; }
#pragma unroll
        for (int mi = 0; mi < MT; ++mi)
#pragma unroll
            for (int ni = 0; ni < NT; ++ni)
                c[mi][ni] = __builtin_amdgcn_wmma_f32_16x16x4_f32(
                    false, a[mi], false, b[ni], (short)0, c[mi][ni], false, false);
    }

#pragma unroll
    for (int ni = 0; ni < NT; ++ni) {
        const int col = n0 + ni * 16 + lid;
        const float bv = bias ? bias[col] : 0.f;
#pragma unroll
        for (int mi = 0; mi < MT; ++mi) {
#pragma unroll
            for (int r = 0; r < 8; ++r) {
                const int row = m0 + mi * 16 + r + (half << 3);
                float v = c[mi][ni][r] + bv;
                if (act == 1)      v = 0.5f * v * (1.f + erff(v * 0.70710678118654752f));
                else if (act == 2) v = (v > 20.f) ? v : log1pf(expf(v));
                if (resid) v += resid[(size_t)row * ldr + col];
                Y[(size_t)row * ldc + ycol0 + col] = v;
            }
        }
    }
}

// =====================================================================
// LayerNorm over last dim (C == DM == blockDim.x == 256)
// =====================================================================
__global__ void ln_kernel(const float* __restrict__ in,
                          const float* __restrict__ gg,
                          const float* __restrict__ bb,
                          float* __restrict__ out)
{
    const int row = blockIdx.x;
    const int tid = threadIdx.x;
    const float v = in[(size_t)row * DM + tid];
    __shared__ float red[DM];
    red[tid] = v; __syncthreads();
    for (int s = 128; s > 0; s >>= 1) { if (tid < s) red[tid] += red[tid + s]; __syncthreads(); }
    const float mu = red[0] * (1.f / DM);
    __syncthreads();
    const float d = v - mu;
    red[tid] = d * d; __syncthreads();
    for (int s = 128; s > 0; s >>= 1) { if (tid < s) red[tid] += red[tid + s]; __syncthreads(); }
    const float var = red[0] * (1.f / DM);
    out[(size_t)row * DM + tid] = d * rsqrtf(var + EPSV) * gg[tid] + bb[tid];
}

// =====================================================================
// Mamba causal depthwise conv (K=4, left pad 3) + bias + SiLU
// x_in lives in xz columns [0, DI) with row stride 2*DI.
// =====================================================================
__global__ void conv_silu_kernel(const float* __restrict__ xz,
                                 const float* __restrict__ w,   // (DI,1,4)
                                 const float* __restrict__ bias,
                                 float* __restrict__ u, int total)
{
    const int i = blockIdx.x * blockDim.x + threadIdx.x;
    if (i >= total) return;
    const int c = i % DI;
    const int l = (i / DI) % LL;
    const int b = i / (DI * LL);
    float s = bias[c];
#pragma unroll
    for (int j = 0; j < 4; ++j) {
        const int ls = l - 3 + j;
        if (ls >= 0) s += w[c * 4 + j] * xz[(size_t)(b * LL + ls) * (2 * DI) + c];
    }
    u[i] = s * sigf(s);
}

// =====================================================================
// Selective scan. grid = (BB, DI/64), 64 threads per block (2 waves),
// spread across WGPs. 16-state recurrence in registers per channel.
// Fused epilogue: yg = (y + u*D) * silu(z)
// =====================================================================
__global__ void scan_kernel(const float* __restrict__ u,
                            const float* __restrict__ dt,
                            const float* __restrict__ dbl,   // (M,48): [dtr | B | C]
                            const float* __restrict__ xz,    // z at cols [DI,2DI)
                            const float* __restrict__ A_log, // (DI,DS)
                            const float* __restrict__ Dp,    // (DI,)
                            float* __restrict__ yg)
{
    const int b = blockIdx.x;
    const int d = blockIdx.y * 64 + threadIdx.x;
    float a[DS], h[DS];
#pragma unroll
    for (int n = 0; n < DS; ++n) { a[n] = -expf(A_log[d * DS + n]); h[n] = 0.f; }
    const float Dd = Dp[d];
    __shared__ float BC[2 * DS];
    for (int t = 0; t < LL; ++t) {
        const int row = b * LL + t;
        if (threadIdx.x < 2 * DS) BC[threadIdx.x] = dbl[(size_t)row * 48 + DTRK + threadIdx.x];
        __syncthreads();
        const float dtv = dt[(size_t)row * DI + d];
        const float uv  = u[(size_t)row * DI + d];
        const float du  = dtv * uv;
        float y = 0.f;
#pragma unroll
        for (int n = 0; n < DS; ++n) {
            h[n] = expf(dtv * a[n]) * h[n] + du * BC[n];
            y += h[n] * BC[DS + n];
        }
        const float zv = xz[(size_t)row * (2 * DI) + DI + d];
        yg[(size_t)row * DI + d] = (y + uv * Dd) * (zv * sigf(zv));
        __syncthreads();
    }
}

// ---------------- small elementwise helpers ----------------
__global__ void flip_kernel(const float* __restrict__ in, float* __restrict__ out, int total)
{
    const int i = blockIdx.x * blockDim.x + threadIdx.x;
    if (i >= total) return;
    const int c = i % DM, l = (i / DM) % LL, b = i / (DM * LL);
    out[i] = in[(size_t)(b * LL + (LL - 1 - l)) * DM + c];
}

__global__ void addflip_kernel(float* __restrict__ acc, const float* __restrict__ ym, int total)
{
    const int i = blockIdx.x * blockDim.x + threadIdx.x;
    if (i >= total) return;
    const int c = i % DM, l = (i / DM) % LL, b = i / (DM * LL);
    acc[i] += ym[(size_t)(b * LL + (LL - 1 - l)) * DM + c];
}

__global__ void resid_mean_kernel(const float* __restrict__ base,
                                  const float* __restrict__ acc,
                                  float* __restrict__ g, int total)
{
    const int i = blockIdx.x * blockDim.x + threadIdx.x;
    if (i >= total) return;
    g[i] = base[i] + 0.5f * acc[i];
}

__global__ void copycat_kernel(const float* __restrict__ g, float* __restrict__ cat, int total)
{
    const int i = blockIdx.x * blockDim.x + threadIdx.x;
    if (i >= total) return;
    cat[(size_t)(i / DM) * (3 * DM) + (i % DM)] = g[i];
}

__global__ void zero8_kernel(float* s) { if (threadIdx.x < 8) s[threadIdx.x] = 0.f; }

// GroupNorm(1,C) stats over (C,L) per batch: sum & sumsq via atomics
__global__ void gn_stats_kernel(const float* __restrict__ in, float* __restrict__ stats, int perB)
{
    const int b = blockIdx.y;
    const float* p = in + (size_t)b * perB;
    float s = 0.f, q = 0.f;
    for (int i = blockIdx.x * blockDim.x + threadIdx.x; i < perB; i += gridDim.x * blockDim.x) {
        const float v = p[i]; s += v; q += v * v;
    }
    __shared__ float rs[256], rq[256];
    const int tid = threadIdx.x;
    rs[tid] = s; rq[tid] = q; __syncthreads();
    for (int st = 128; st > 0; st >>= 1) {
        if (tid < st) { rs[tid] += rs[tid + st]; rq[tid] += rq[tid + st]; }
        __syncthreads();
    }
    if (tid == 0) { atomicAdd(&stats[b * 2 + 0], rs[0]); atomicAdd(&stats[b * 2 + 1], rq[0]); }
}

__global__ void gn_silu_kernel(const float* __restrict__ in, const float* __restrict__ stats,
                               const float* __restrict__ gg, const float* __restrict__ bb,
                               float* __restrict__ out, int total)
{
    const int i = blockIdx.x * blockDim.x + threadIdx.x;
    if (i >= total) return;
    const int c = i % DM;
    const int b = i / (LL * DM);
    const float inv = 1.f / (float)(LL * DM);
    const float mu  = stats[b * 2 + 0] * inv;
    const float var = stats[b * 2 + 1] * inv - mu * mu;
    const float v = (in[i] - mu) * rsqrtf(var + EPSV) * gg[c] + bb[c];
    out[i] = v * sigf(v);
}

// same-pad depthwise conv along L (k = 5 or 7)
__global__ void dwconv_kernel(const float* __restrict__ in, const float* __restrict__ w,
                              float* __restrict__ out, int k, int total)
{
    const int i = blockIdx.x * blockDim.x + threadIdx.x;
    if (i >= total) return;
    const int c = i % DM, l = (i / DM) % LL, b = i / (DM * LL);
    const int pad = k / 2;
    float s = 0.f;
    for (int j = 0; j < k; ++j) {
        const int ls = l + j - pad;
        if (ls >= 0 && ls < LL) s += w[c * k + j] * in[(size_t)(b * LL + ls) * DM + c];
    }
    out[i] = s;
}

// =====================================================================
// Host orchestration
// =====================================================================
extern "C" void kernel_launch(void* const* d_in, const int* in_sizes, int n_in,
                              void* d_out, int out_size, void* d_ws, size_t ws_size,
                              hipStream_t stream)
{
    (void)in_sizes; (void)n_in; (void)out_size; (void)ws_size;
    const float* x = (const float*)d_in[0];
    auto F = [&](int i) { return (const float*)d_in[i]; };

    // workspace carve-up
    float* ws = (float*)d_ws;
    size_t off = 0;
    auto carve = [&](size_t n) { float* p = ws + off; off += n; return p; };
    float* g    = carve((size_t)MROWS * DM);        // residual stream
    float* xn   = carve((size_t)MROWS * DM);        // LN output / local h1
    float* xrev = carve((size_t)MROWS * DM);        // flipped input / local conv out
    float* xz   = carve((size_t)MROWS * 2 * DI);    // in_proj out (x_in|z) / FFN hidden
    float* ub   = carve((size_t)MROWS * DI);        // conv+silu output u
    float* dblb = carve((size_t)MROWS * 48);        // x_proj out [dtr|B|C]
    float* dtb  = carve((size_t)MROWS * DI);        // softplus dt
    float* ygb  = carve((size_t)MROWS * DI);        // gated scan output
    float* ym   = carve((size_t)MROWS * DM);        // bwd mamba output (pre-flip)
    float* acc  = carve((size_t)MROWS * DM);        // y_fwd (+ flipped y_bwd)
    float* cat  = carve((size_t)MROWS * 3 * DM);    // concat features
    float* stats = carve(8);

    const dim3 blk32(32), blk256(256);
    auto nb = [](size_t n) { return (unsigned)((n + 255) / 256); };
    const int totDM = MROWS * DM;
    const int totDI = MROWS * DI;

    auto gemm = [&](const float* X, int lda, const float* W, int ldw,
                    const float* bias, const float* resid, int ldr,
                    float* Y, int ldc, int ycol0, int N, int K, int act) {
        if ((N & 31) == 0) {
            dim3 grid(N / 32, MROWS / 32);
            gemm_wmma_blk<2, 2><<<grid, blk32, 0, stream>>>(
                X, lda, W, ldw, bias, resid, ldr, Y, ldc, ycol0, K, act);
        } else {
            dim3 grid(N / 16, MROWS / 16);
            gemm_wmma_blk<1, 1><<<grid, blk32, 0, stream>>>(
                X, lda, W, ldw, bias, resid, ldr, Y, ldc, ycol0, K, act);
        }
    };

    // one mamba direction; params at d_in[pb..pb+8]:
    // in_proj, conv_w, conv_b, x_proj, dt_w, dt_b, A_log, D, out_proj
    auto run_mamba = [&](int pb, const float* xin, float* yout) {
        gemm(xin, DM, F(pb + 0), DM, nullptr, nullptr, 0, xz, 2 * DI, 0, 2 * DI, DM, 0);
        conv_silu_kernel<<<nb(totDI), blk256, 0, stream>>>(xz, F(pb + 1), F(pb + 2), ub, totDI);
        gemm(ub, DI, F(pb + 3), DI, nullptr, nullptr, 0, dblb, 48, 0, 48, DI, 0);
        gemm(dblb, 48, F(pb + 4), DTRK, F(pb + 5), nullptr, 0, dtb, DI, 0, DI, DTRK, 2);
        scan_kernel<<<dim3(BB, DI / 64), dim3(64), 0, stream>>>(ub, dtb, dblb, xz,
                                                                F(pb + 6), F(pb + 7), ygb);
        gemm(ygb, DI, F(pb + 8), DI, nullptr, nullptr, 0, yout, DM, 0, DM, DI, 0);
    };

    // ---------------- bimamba layers ----------------
    const float* cur = x;
    for (int layer = 0; layer < 2; ++layer) {
        const int base = 1 + layer * 26;
        ln_kernel<<<dim3(MROWS), dim3(DM), 0, stream>>>(cur, F(base + 0), F(base + 1), xn);
        // forward direction -> acc
        run_mamba(base + 2, xn, acc);
        // backward direction on flipped input
        flip_kernel<<<nb(totDM), blk256, 0, stream>>>(xn, xrev, totDM);
        run_mamba(base + 11, xrev, ym);
        addflip_kernel<<<nb(totDM), blk256, 0, stream>>>(acc, ym, totDM);
        // x = x + 0.5*(y_fwd + y_bwd)
        resid_mean_kernel<<<nb(totDM), blk256, 0, stream>>>(cur, acc, g, totDM);
        // FFN
        ln_kernel<<<dim3(MROWS), dim3(DM), 0, stream>>>(g, F(base + 20), F(base + 21), xn);
        gemm(xn, DM, F(base + 22), DM, F(base + 23), nullptr, 0, xz, 4 * DM, 0, 4 * DM, DM, 1);
        gemm(xz, 4 * DM, F(base + 24), 4 * DM, F(base + 25), g, DM, g, DM, 0, DM, 4 * DM, 0);
        cur = g;
    }

    // g -> concat columns [0, DM)
    copycat_kernel<<<nb(totDM), blk256, 0, stream>>>(g, cat, totDM);

    // ---------------- local conv branches ----------------
    for (int j = 0; j < 2; ++j) {
        const int lb = 53 + j * 9;   // w1,b1,gn1_g,gn1_b,dw,gn2_g,gn2_b,w2,b2
        const int k = (j == 0) ? 5 : 7;
        gemm(x, DM, F(lb + 0), DM, F(lb + 1), nullptr, 0, xn, DM, 0, DM, DM, 0);
        zero8_kernel<<<1, 8, 0, stream>>>(stats);
        gn_stats_kernel<<<dim3(64, BB), blk256, 0, stream>>>(xn, stats, LL * DM);
        gn_silu_kernel<<<nb(totDM), blk256, 0, stream>>>(xn, stats, F(lb + 2), F(lb + 3), acc, totDM);
        dwconv_kernel<<<nb(totDM), blk256, 0, stream>>>(acc, F(lb + 4), xrev, k, totDM);
        zero8_kernel<<<1, 8, 0, stream>>>(stats);
        gn_stats_kernel<<<dim3(64, BB), blk256, 0, stream>>>(xrev, stats, LL * DM);
        gn_silu_kernel<<<nb(totDM), blk256, 0, stream>>>(xrev, stats, F(lb + 5), F(lb + 6), acc, totDM);
        gemm(acc, DM, F(lb + 7), DM, F(lb + 8), nullptr, 0, cat, 3 * DM, DM * (1 + j), DM, DM, 0);
    }

    // ---------------- final projection + lss residual ----------------
    gemm(cat, 3 * DM, F(71), 3 * DM, F(72), x, DM, (float*)d_out, DM, 0, DM, 3 * DM, 0);
}